// SamplePredictor_56075093016589
// MI455X (gfx1250) — compile-verified
//
#include <hip/hip_runtime.h>
#include <hip/hip_bf16.h>

typedef __attribute__((ext_vector_type(2))) float v2f;
typedef __attribute__((ext_vector_type(8))) float v8f;

// ---------------- utility ----------------
__global__ void k_zero(float* __restrict__ p, long long n) {
    long long i = (long long)blockIdx.x * blockDim.x + threadIdx.x;
    if (i < n) p[i] = 0.0f;
}

// ---------------- degree / normalization ----------------
__global__ void k_degree(const long long* __restrict__ ei, float* __restrict__ deg, int E) {
    int e = blockIdx.x * blockDim.x + threadIdx.x;
    if (e < E) {
        int d = (int)ei[(long long)E + e];   // dst row of edge_index
        atomicAdd(&deg[d], 1.0f);
    }
}

__global__ void k_dis(const float* __restrict__ deg, float* __restrict__ dis, int n) {
    int i = blockIdx.x * blockDim.x + threadIdx.x;
    if (i < n) dis[i] = rsqrtf(deg[i] + 1.0f);   // deg includes +1 self-loop
}

// ---------------- layer-1 dense: xw = x @ W1  (K=5) ----------------
__global__ void k_xw1(const float* __restrict__ x, const float* __restrict__ W1,
                      float* __restrict__ xw, int n) {
    int idx = blockIdx.x * blockDim.x + threadIdx.x;   // node*64 + o
    int node = idx >> 6;
    int o = idx & 63;
    if (node < n) {
        const float* xr = x + (long long)node * 5;
        float s = 0.0f;
#pragma unroll
        for (int k = 0; k < 5; ++k) s += xr[k] * W1[k * 64 + o];
        xw[idx] = s;
    }
}

// ---------------- edge aggregation: agg[dst] += dis[src]*dis[dst]*feat[src] ----------------
__global__ void k_agg(const float* __restrict__ feat, const long long* __restrict__ ei,
                      const float* __restrict__ dis, float* __restrict__ agg, int E) {
    long long idx = (long long)blockIdx.x * blockDim.x + threadIdx.x;
    long long e = idx >> 6;
    int f = (int)(idx & 63);
    if (e < E) {
        int s = (int)ei[e];
        int d = (int)ei[(long long)E + e];
        float norm = dis[s] * dis[d];
        atomicAdd(&agg[(long long)d * 64 + f], norm * feat[(long long)s * 64 + f]);
    }
}

// ---------------- finish: h = relu(agg + (1/deg)*xw + b), in place in agg ----------------
__global__ void k_finish(const float* __restrict__ xw, const float* __restrict__ dis,
                         const float* __restrict__ bias, float* __restrict__ agg_h, int n) {
    long long idx = (long long)blockIdx.x * blockDim.x + threadIdx.x;
    int node = (int)(idx >> 6);
    int f = (int)(idx & 63);
    if (node < n) {
        float di = dis[node];
        float invdeg = di * di;                  // 1/deg
        float v = agg_h[idx] + invdeg * xw[idx] + bias[f];
        agg_h[idx] = fmaxf(v, 0.0f);
    }
}

// ---------------- layer-2 dense GEMM via FP32 WMMA: D[n,64] = A[n,64] @ B[64,64] ----------
__global__ void __launch_bounds__(256) k_gemm_wmma(const float* __restrict__ A,
                                                   const float* __restrict__ B,
                                                   float* __restrict__ D, int n) {
    __shared__ float Bs[64 * 64];
    for (int i = threadIdx.x; i < 64 * 64; i += 256) Bs[i] = B[i];
    __syncthreads();

    int wave = threadIdx.x >> 5;              // 8 waves per block
    int lane = threadIdx.x & 31;
    int gw = blockIdx.x * 8 + wave;           // global wave id
    int mt = gw >> 2;                         // 16-row tile index
    int nt = gw & 3;                          // 16-col tile index (64 cols)
    if (mt * 16 >= n) return;                 // whole-wave uniform: EXEC stays all-ones

    int l  = lane & 15;
    int hi = (lane >> 4) & 1;                 // lanes 16..31 hold K+2 / M+8 halves
    const float* Arow = A + (long long)(mt * 16 + l) * 64;
    int n0 = nt * 16;

    v8f c = {};
#pragma unroll
    for (int kk = 0; kk < 64; kk += 4) {
        int k2 = kk + hi * 2;
        // A 16x4 fragment: lanes 0-15 -> (M=l, K=kk..kk+1); lanes 16-31 -> K=kk+2..kk+3
        v2f a = *(const v2f*)(Arow + k2);
        // B 4x16 fragment: row K striped across lanes within a VGPR
        v2f b;
        b.x = Bs[k2 * 64 + n0 + l];
        b.y = Bs[(k2 + 1) * 64 + n0 + l];
        c = __builtin_amdgcn_wmma_f32_16x16x4_f32(false, a, false, b, (short)0, c, false, false);
    }
    // C/D layout: VGPR r -> M = r (lanes 0-15) / M = 8+r (lanes 16-31), N = l
    int mbase = mt * 16 + hi * 8;
#pragma unroll
    for (int r = 0; r < 8; ++r)
        D[(long long)(mbase + r) * 64 + n0 + l] = c[r];
}

// ---------------- mean-pool column sums ----------------
__global__ void k_reduce_embed(const float* __restrict__ h, float* __restrict__ accum, int n) {
    __shared__ float sh[256];
    int t = threadIdx.x;
    int f = t & 63;
    int r4 = t >> 6;                          // 0..3 rows handled per block pass
    float s = 0.0f;
    for (int row = blockIdx.x * 4 + r4; row < n; row += gridDim.x * 4)
        s += h[(long long)row * 64 + f];
    sh[t] = s;
    __syncthreads();
    if (t < 64) {
        float v = sh[t] + sh[t + 64] + sh[t + 128] + sh[t + 192];
        atomicAdd(&accum[f], v);
    }
}

// ---------------- global-feature reductions over x ----------------
__global__ void k_reduce_global(const float* __restrict__ x, float* __restrict__ accum, int n) {
    __shared__ float sh[256];
    int i = blockIdx.x * blockDim.x + threadIdx.x;
    float v[6] = {0, 0, 0, 0, 0, 0};
    if (i < n) {
        float lam = x[(long long)i * 5 + 0];
        float mu  = x[(long long)i * 5 + 1];
        float c   = x[(long long)i * 5 + 2];
        float a   = x[(long long)i * 5 + 3];
        float o   = x[(long long)i * 5 + 4];
        float m = (c == 1.0f) ? 1.0f : 0.0f;
        v[0] = c; v[1] = a; v[2] = o; v[3] = m; v[4] = lam * m; v[5] = mu * m;
    }
#pragma unroll
    for (int q = 0; q < 6; ++q) {
        sh[threadIdx.x] = v[q];
        __syncthreads();
        for (int s = 128; s > 0; s >>= 1) {
            if ((int)threadIdx.x < s) sh[threadIdx.x] += sh[threadIdx.x + s];
            __syncthreads();
        }
        if (threadIdx.x == 0) atomicAdd(&accum[64 + q], sh[0]);
        __syncthreads();
    }
}

// ---------------- final MLP head (single block) ----------------
__global__ void k_final(const float* __restrict__ accum,
                        const float* __restrict__ Wp1, const float* __restrict__ bp1,
                        const float* __restrict__ Wp2, const float* __restrict__ bp2,
                        const int* __restrict__ pT, const int* __restrict__ pTmax,
                        float* __restrict__ out, int n) {
    __shared__ float e[71];
    __shared__ float hid[32];
    int t = threadIdx.x;
    if (t < 64) e[t] = accum[t] / (float)n;                    // global mean pool
    if (t == 64) e[64] = accum[64];                            // n_comp
    if (t == 65) e[65] = accum[65];                            // n_AND
    if (t == 66) e[66] = accum[66];                            // n_OR
    if (t == 67) e[67] = accum[65] + accum[66];                // depth
    if (t == 68) { float ms = accum[67]; e[68] = (ms > 0.0f) ? accum[68] / fmaxf(ms, 1.0f) : 0.0f; }
    if (t == 69) { float ms = accum[67]; e[69] = (ms > 0.0f) ? accum[69] / fmaxf(ms, 1.0f) : 0.0f; }
    if (t == 70) e[70] = (float)pT[0] / (float)pTmax[0];       // T_norm
    __syncthreads();
    if (t < 32) {
        float s = bp1[t];
        for (int i = 0; i < 71; ++i) s += e[i] * Wp1[i * 32 + t];
        hid[t] = fmaxf(s, 0.0f);
    }
    __syncthreads();
    if (t < 2) {
        float s = bp2[t];
        for (int j = 0; j < 32; ++j) s += hid[j] * Wp2[j * 2 + t];
        out[t] = 2.0f + 4.0f / (1.0f + expf(-s));
    }
}

extern "C" void kernel_launch(void* const* d_in, const int* in_sizes, int n_in,
                              void* d_out, int out_size, void* d_ws, size_t ws_size,
                              hipStream_t stream) {
    const float*     x   = (const float*)d_in[0];
    const long long* ei  = (const long long*)d_in[1];   // int64 edge_index [2,E]
    const float*     W1  = (const float*)d_in[2];
    const float*     b1  = (const float*)d_in[3];
    const float*     W2  = (const float*)d_in[4];
    const float*     b2  = (const float*)d_in[5];
    const float*     Wp1 = (const float*)d_in[6];
    const float*     bp1 = (const float*)d_in[7];
    const float*     Wp2 = (const float*)d_in[8];
    const float*     bp2 = (const float*)d_in[9];
    const int*       pT  = (const int*)d_in[10];
    const int*       pTm = (const int*)d_in[11];
    float* out = (float*)d_out;

    int n = in_sizes[0] / 5;
    int E = in_sizes[1] / 2;
    long long nf = (long long)n * 64;

    // workspace layout (fp32): bufA[n*64] | bufB[n*64] | deg[n] | dis[n] | accum[72]
    float* bufA  = (float*)d_ws;
    float* bufB  = bufA + nf;
    float* deg   = bufB + nf;
    float* dis   = deg + n;
    float* accum = dis + n;

    const int B = 256;
    long long eThreads = (long long)E * 64;

    // zero agg(bufB) + deg (contiguous), and accumulators
    long long z1 = nf + n;
    k_zero<<<(unsigned)((z1 + B - 1) / B), B, 0, stream>>>(bufB, z1);
    k_zero<<<1, 128, 0, stream>>>(accum, 72);

    // degrees + normalization
    k_degree<<<(E + B - 1) / B, B, 0, stream>>>(ei, deg, E);
    k_dis<<<(n + B - 1) / B, B, 0, stream>>>(deg, dis, n);

    // layer 1: xw1 = x @ W1 ; aggregate ; finish (h1 in bufB)
    k_xw1<<<(unsigned)((nf + B - 1) / B), B, 0, stream>>>(x, W1, bufA, n);
    k_agg<<<(unsigned)((eThreads + B - 1) / B), B, 0, stream>>>(bufA, ei, dis, bufB, E);
    k_finish<<<(unsigned)((nf + B - 1) / B), B, 0, stream>>>(bufA, dis, b1, bufB, n);

    // layer 2: xw2 = h1 @ W2 via FP32 WMMA (writes bufA; xw1 dead)
    int mtiles = (n + 15) / 16;
    int waves = mtiles * 4;                    // 4 N-tiles of 16 over 64 cols
    k_gemm_wmma<<<(waves + 7) / 8, B, 0, stream>>>(bufB, W2, bufA, n);

    // re-zero agg, aggregate, finish (h2 in bufB)
    k_zero<<<(unsigned)((nf + B - 1) / B), B, 0, stream>>>(bufB, nf);
    k_agg<<<(unsigned)((eThreads + B - 1) / B), B, 0, stream>>>(bufA, ei, dis, bufB, E);
    k_finish<<<(unsigned)((nf + B - 1) / B), B, 0, stream>>>(bufA, dis, b2, bufB, n);

    // pooling + global features + head
    k_reduce_embed<<<256, B, 0, stream>>>(bufB, accum, n);
    k_reduce_global<<<(n + B - 1) / B, B, 0, stream>>>(x, accum, n);
    k_final<<<1, 128, 0, stream>>>(accum, Wp1, bp1, Wp2, bp2, pT, pTm, out, n);
}